// VSN_85925115723999
// MI455X (gfx1250) — compile-verified
//
#include <hip/hip_runtime.h>
#include <hip/hip_bf16.h>

// B=8, A=8, T=128, F=64, H=128
#define BA   64      // B*A
#define TT   128
#define FF   64
#define HH   128
#define LDSTRIDE 132 // 128 + 4 pad -> conflict-free strided A-frag reads

typedef __attribute__((ext_vector_type(2))) float v2f;
typedef __attribute__((ext_vector_type(8))) float v8f;

__device__ __forceinline__ float eluf(float v) {
    return v > 0.0f ? v : (__expf(v) - 1.0f);
}

// ---------------------------------------------------------------------------
// Kernel 1: gating logits  g = elu(x@w1 + s@w2 + b1 + b2) @ w3 + b3
// One (ba,t) row per block, 128 threads. Writes transposed [ba, f, t] so the
// time-axis softmax and kernel-3 gate reads are contiguous.
// ---------------------------------------------------------------------------
__global__ __launch_bounds__(128) void vsn_gating(
    const float* __restrict__ x, const float* __restrict__ s,
    const float* __restrict__ w1, const float* __restrict__ b1,
    const float* __restrict__ w2, const float* __restrict__ b2,
    const float* __restrict__ w3, const float* __restrict__ b3,
    float* __restrict__ logits) {
    __shared__ float g1[HH];
    const int row = blockIdx.x;          // ba*T + t
    const int ba  = row >> 7;
    const int t   = row & (TT - 1);
    const int h   = threadIdx.x;         // 0..127
    const float* xr = x + (size_t)row * FF;
    const float* sr = s + (size_t)row * FF;
    float acc = b1[h] + b2[h];
#pragma unroll 8
    for (int k = 0; k < FF; ++k)
        acc += xr[k] * w1[k * HH + h] + sr[k] * w2[k * HH + h];
    g1[h] = eluf(acc);
    __syncthreads();
    if (h < FF) {
        float a2 = b3[h];
#pragma unroll 8
        for (int j = 0; j < HH; ++j)
            a2 += g1[j] * w3[j * FF + h];
        logits[(ba * FF + h) * TT + t] = a2;
    }
}

// ---------------------------------------------------------------------------
// Kernel 2: in-place softmax over the T axis. One wave32 per 128-long row.
// ---------------------------------------------------------------------------
__global__ __launch_bounds__(256) void vsn_softmax_t(float* __restrict__ logits) {
    const int wv   = threadIdx.x >> 5;
    const int lane = threadIdx.x & 31;
    const int row  = blockIdx.x * 8 + wv;      // 0 .. BA*FF-1
    float* p = logits + (size_t)row * TT + lane * 4;
    float4 v = *(const float4*)p;
    float m = fmaxf(fmaxf(v.x, v.y), fmaxf(v.z, v.w));
#pragma unroll
    for (int off = 16; off > 0; off >>= 1)
        m = fmaxf(m, __shfl_xor(m, off, 32));
    v.x = __expf(v.x - m); v.y = __expf(v.y - m);
    v.z = __expf(v.z - m); v.w = __expf(v.w - m);
    float ssum = v.x + v.y + v.z + v.w;
#pragma unroll
    for (int off = 16; off > 0; off >>= 1)
        ssum += __shfl_xor(ssum, off, 32);
    const float inv = 1.0f / ssum;
    v.x *= inv; v.y *= inv; v.z *= inv; v.w *= inv;
    *(float4*)p = v;
}

// ---------------------------------------------------------------------------
// Kernel 3 (dominant, WMMA): out[t,k] = sum_f wt[t,f]*( elu(x*jw1+jb1)@jw3[f] )
//                                     + (wt @ jb3)
// Grid: 128 blocks = (ba, 64-row half of T). 8 waves; wave w owns N-strip
// n0 = 16*w of the H=128 output columns, 4 M-tiles of 16 rows.
// Gate wt is folded into the A rows, so accumulation is a pure GEMM chain:
// 64 f-steps x 32 K-steps x 4 tiles of v_wmma_f32_16x16x4_f32.
// ---------------------------------------------------------------------------
__global__ __launch_bounds__(256) void vsn_main(
    const float* __restrict__ x,   const float* __restrict__ wt,
    const float* __restrict__ jw1, const float* __restrict__ jb1,
    const float* __restrict__ jw3, const float* __restrict__ jb3,
    float* __restrict__ out) {
    __shared__ float Hs[64 * LDSTRIDE];     // 33,792 B

    const int ba = blockIdx.x >> 1;
    const int mh = blockIdx.x & 1;          // which 64-row half of T
    const int tbase = mh * 64;

    const int lane  = threadIdx.x & 31;
    const int wv    = threadIdx.x >> 5;
    const int lhalf = lane >> 4;            // 0: K,K+1 ; 1: K+2,K+3
    const int l15   = lane & 15;
    const int n     = wv * 16 + l15;        // this lane's output column

    v8f acc[4] = {};

    const float* xba  = x  + (size_t)(ba * TT + tbase) * FF;
    const float* wtba = wt + (size_t)ba * FF * TT;

    for (int f = 0; f < FF; ++f) {
        // ---- build gated A tile: Hs[row][j] = wt[t,f] * elu(x[t,f]*jw1[f,j]+jb1[f,j])
        const float* jw1f = jw1 + f * HH;
        const float* jb1f = jb1 + f * HH;
        const float* wtf  = wtba + f * TT + tbase;
#pragma unroll
        for (int i4 = threadIdx.x; i4 < 64 * (HH / 4); i4 += 256) {
            const int row = i4 >> 5;            // 0..63
            const int jj  = (i4 & 31) * 4;      // 0..124
            const float xv  = xba[row * FF + f];
            const float wtv = wtf[row];
            const float4 wrow = *(const float4*)(jw1f + jj);
            const float4 brow = *(const float4*)(jb1f + jj);
            float4 hv;
            hv.x = wtv * eluf(fmaf(xv, wrow.x, brow.x));
            hv.y = wtv * eluf(fmaf(xv, wrow.y, brow.y));
            hv.z = wtv * eluf(fmaf(xv, wrow.z, brow.z));
            hv.w = wtv * eluf(fmaf(xv, wrow.w, brow.w));
            *(float4*)&Hs[row * LDSTRIDE + jj] = hv;
        }
        __syncthreads();

        // ---- [64x128] @ jw3[f] (128x128), B frags straight from L2-resident jw3
        const float* jw3f = jw3 + (size_t)f * HH * HH;
#pragma unroll 2
        for (int kk = 0; kk < 32; ++kk) {
            const int kf = kk * 4 + 2 * lhalf;
            v2f bfrag;
            bfrag.x = jw3f[kf * HH + n];
            bfrag.y = jw3f[(kf + 1) * HH + n];
#pragma unroll
            for (int mt = 0; mt < 4; ++mt) {
                const v2f afrag = *(const v2f*)&Hs[(mt * 16 + l15) * LDSTRIDE + kf];
                acc[mt] = __builtin_amdgcn_wmma_f32_16x16x4_f32(
                    false, afrag, false, bfrag, (short)0, acc[mt], false, false);
            }
        }
        __syncthreads();   // protect Hs before next f overwrites it
    }

    // ---- bias term: acc += wt[t,:] @ jb3   (K = F = 64)
#pragma unroll 2
    for (int kk = 0; kk < 16; ++kk) {
        const int kf = kk * 4 + 2 * lhalf;   // feature index
        v2f bfrag;
        bfrag.x = jb3[kf * HH + n];
        bfrag.y = jb3[(kf + 1) * HH + n];
#pragma unroll
        for (int mt = 0; mt < 4; ++mt) {
            const int t = tbase + mt * 16 + l15;
            v2f afrag;
            afrag.x = wtba[kf * TT + t];
            afrag.y = wtba[(kf + 1) * TT + t];
            acc[mt] = __builtin_amdgcn_wmma_f32_16x16x4_f32(
                false, afrag, false, bfrag, (short)0, acc[mt], false, false);
        }
    }

    // ---- store C tiles: VGPR v -> rows v (lanes 0-15) / v+8 (lanes 16-31)
#pragma unroll
    for (int mt = 0; mt < 4; ++mt)
#pragma unroll
        for (int v = 0; v < 8; ++v) {
            const int t = tbase + mt * 16 + v + 8 * lhalf;
            out[(size_t)(ba * TT + t) * HH + n] = acc[mt][v];
        }
}

// ---------------------------------------------------------------------------
extern "C" void kernel_launch(void* const* d_in, const int* in_sizes, int n_in,
                              void* d_out, int out_size, void* d_ws, size_t ws_size,
                              hipStream_t stream) {
    const float* x   = (const float*)d_in[0];
    const float* s   = (const float*)d_in[1];
    const float* w1  = (const float*)d_in[2];
    const float* b1  = (const float*)d_in[3];
    const float* w2  = (const float*)d_in[4];
    const float* b2  = (const float*)d_in[5];
    const float* w3  = (const float*)d_in[6];
    const float* b3  = (const float*)d_in[7];
    const float* jw1 = (const float*)d_in[8];
    const float* jb1 = (const float*)d_in[9];
    const float* jw3 = (const float*)d_in[10];
    const float* jb3 = (const float*)d_in[11];
    float* out = (float*)d_out;

    float* wt = (float*)d_ws;   // [ba, f, t] = 64*64*128 floats = 2 MB, fully rewritten per call

    vsn_gating  <<<BA * TT, 128, 0, stream>>>(x, s, w1, b1, w2, b2, w3, b3, wt);
    vsn_softmax_t<<<(BA * FF) / 8, 256, 0, stream>>>(wt);
    vsn_main    <<<BA * 2, 256, 0, stream>>>(x, wt, jw1, jb1, jw3, jb3, out);
}